// EigenMatrixGenerator_42657615184237
// MI455X (gfx1250) — compile-verified
//
#include <hip/hip_runtime.h>
#include <math.h>

namespace {

constexpr int    G   = 1600;                         // g = N*N
constexpr int    S3  = 4800;                         // s = 3g
constexpr size_t SS  = (size_t)S3 * (size_t)S3;      // 23,040,000 elements per matrix
constexpr size_t N16 = (2 * SS) / 4;                 // float4 chunks covering A ++ W = 11,520,000

// Fill geometry: each block covers 2048 contiguous 16B chunks (32 KB);
// 256 threads x 8 chunks each, addressed with immediate offsets.
constexpr int    CHUNKS_PER_BLOCK = 2048;
constexpr int    FILL_BLOCKS      = (int)(N16 / CHUNKS_PER_BLOCK);   // 5625
static_assert((size_t)FILL_BLOCKS * CHUNKS_PER_BLOCK == N16, "exact tiling");

struct Cf { float r, i; };
__device__ __forceinline__ Cf cmul(Cf a, Cf b) { return { a.r*b.r - a.i*b.i, a.r*b.i + a.i*b.r }; }
__device__ __forceinline__ Cf cadd(Cf a, Cf b) { return { a.r + b.r, a.i + b.i }; }
__device__ __forceinline__ Cf cconj(Cf a)      { return { a.r, -a.i }; }

// jax.nn.softplus(x) = max(x,0) + log1p(exp(-|x|))  (stable for x up to ~306 here)
__device__ __forceinline__ float softplusf(float x) {
  return fmaxf(x, 0.0f) + log1pf(expf(-fabsf(x)));
}

} // namespace

#if defined(__gfx1250__) && __has_builtin(__builtin_amdgcn_global_store_async_from_lds_b128)
#define USE_ASYNC_STORE 1
typedef int v4i __attribute__((ext_vector_type(4)));
typedef __attribute__((address_space(1))) v4i* glb_v4i_p;   // global (prints as "__device__ *")
typedef __attribute__((address_space(3))) v4i* lds_v4i_p;   // LDS (shared)
#endif

// ---------------------------------------------------------------------------
// Kernel 1: stream 184.3 MB of zeros over the A and W regions of d_out.
// CDNA5 path: GLOBAL_STORE_ASYNC_FROM_LDS_B128 (ASYNCcnt) — data sourced from
// a 512B LDS zero buffer; 8 stores per thread off one base VGPR pair using
// literal immediate offsets, so the steady state is pure async-store issue.
// Per wave instruction: 32 lanes x 16B = one contiguous 512B burst.
// ---------------------------------------------------------------------------
__global__ void __launch_bounds__(256) zero_fill_kernel(float* __restrict__ out) {
  const size_t ci = (size_t)blockIdx.x * CHUNKS_PER_BLOCK + threadIdx.x;  // first chunk
#if defined(USE_ASYNC_STORE)
  __shared__ v4i zbuf[32];
  if (threadIdx.x < 32) zbuf[threadIdx.x] = (v4i){0, 0, 0, 0};
  __syncthreads();

  lds_v4i_p lsrc = (lds_v4i_p)&zbuf[threadIdx.x & 31];   // distinct 16B per lane
  glb_v4i_p base = (glb_v4i_p)(reinterpret_cast<v4i*>(out) + ci);

  // 8 chunks per thread, stride 256 chunks = 4096 bytes (imm24 offsets)
  __builtin_amdgcn_global_store_async_from_lds_b128(base, lsrc, 0 * 4096, 0);
  __builtin_amdgcn_global_store_async_from_lds_b128(base, lsrc, 1 * 4096, 0);
  __builtin_amdgcn_global_store_async_from_lds_b128(base, lsrc, 2 * 4096, 0);
  __builtin_amdgcn_global_store_async_from_lds_b128(base, lsrc, 3 * 4096, 0);
  __builtin_amdgcn_global_store_async_from_lds_b128(base, lsrc, 4 * 4096, 0);
  __builtin_amdgcn_global_store_async_from_lds_b128(base, lsrc, 5 * 4096, 0);
  __builtin_amdgcn_global_store_async_from_lds_b128(base, lsrc, 6 * 4096, 0);
  __builtin_amdgcn_global_store_async_from_lds_b128(base, lsrc, 7 * 4096, 0);

#if __has_builtin(__builtin_amdgcn_s_wait_asynccnt)
  __builtin_amdgcn_s_wait_asynccnt(0);   // drain ASYNCcnt (S_ENDPGM would also)
#endif
#else
  // Fallback: plain b128 stores at the same addresses
  float4* o4 = reinterpret_cast<float4*>(out);
  const float4 z = make_float4(0.f, 0.f, 0.f, 0.f);
#pragma unroll
  for (int k = 0; k < 8; ++k) o4[ci + (size_t)k * 256] = z;
#endif
}

// ---------------------------------------------------------------------------
// Kernel 2: per grid cell t, compute lam1/lam2/v2 exactly as the reference
// (fp32, same op order) and write the 9 diagonal entries of each 3x3 block
// pair into A and W, plus the c_pos and k tail outputs.
// Runs AFTER zero_fill_kernel on the same stream (overwrites zeros).
// ---------------------------------------------------------------------------
__global__ void eigen_diag_kernel(const float* __restrict__ c_in,
                                  const float* __restrict__ k_in,
                                  const float* __restrict__ kp_in,
                                  float* __restrict__ out) {
  int t = blockIdx.x * blockDim.x + threadIdx.x;
  if (t >= G) return;

  const float xi = 1.57079632679489662f;   // pi/2 (DT = 1)

  float cp  = softplusf(c_in[t]);
  float kx  = softplusf(k_in[t]);
  float kpp = softplusf(kp_in[t]);

  float lam1 = 1.0f / (1.0f + kx);
  float ikp  = 1.0f / (1.0f + kpp);
  float re2  = 0.5f * (ikp + lam1);
  float im2  = cp * xi / sqrtf((1.0f + kpp) * (1.0f + kx));

  float beta = lam1;                        // DT/(1+DT*k_x) with DT=1
  float ar   = (re2 - lam1) + 1e-6f;        // alpha = lam2 - lam1 + 1e-6
  float ai   = im2;
  float den  = ar * ar + ai * ai;
  float bxi  = beta * xi;                   // numerator of v2 is i*(beta*xi)
  Cf v2 = {  bxi * ai / den, bxi * ar / den };
  Cf v3 = { -v2.r, v2.i };                  // -conj(v2)

  const Cf one = { 1.f, 0.f }, zer = { 0.f, 0.f };
  Cf P[3][3] = { { one, one, one },
                 { zer, v2,  v3  },
                 { zer, v2,  v3  } };
  Cf L[3]    = { { lam1, 0.f }, { re2, im2 }, { re2, -im2 } };

#pragma unroll
  for (int i = 0; i < 3; ++i) {
#pragma unroll
    for (int j = 0; j < 3; ++j) {
      Cf acc = zer;                         // sum over the 3 nonzero m terms
#pragma unroll
      for (int m = 0; m < 3; ++m)
        acc = cadd(acc, cmul(cmul(P[i][m], L[m]), cconj(P[j][m])));
      float a = acc.r;

      size_t row = (size_t)i * G + (size_t)t;
      size_t col = (size_t)j * G + (size_t)t;
      size_t idx = row * (size_t)S3 + col;
      out[idx] = a;                                           // A
      out[SS + idx] = (a - (i == j ? 0.97f : 0.0f)) / 0.03f;  // W
    }
  }

  out[2 * SS + t]     = cp;        // c_pos.reshape(n, n)
  out[2 * SS + G + t] = k_in[t];   // k (passthrough)
}

extern "C" void kernel_launch(void* const* d_in, const int* in_sizes, int n_in,
                              void* d_out, int out_size, void* d_ws, size_t ws_size,
                              hipStream_t stream) {
  (void)in_sizes; (void)n_in; (void)d_ws; (void)ws_size; (void)out_size;

  // setup_inputs order: c, k, kp, dx, dy (dx/dy unused by the reference math)
  const float* c  = (const float*)d_in[0];
  const float* k  = (const float*)d_in[1];
  const float* kp = (const float*)d_in[2];
  float* out = (float*)d_out;

  // 1) zero A and W regions (store-bandwidth bound: ~184 MB -> ~7.9 us floor)
  zero_fill_kernel<<<FILL_BLOCKS, 256, 0, stream>>>(out);
  // 2) overwrite block diagonals + tail outputs (ordered by stream)
  eigen_diag_kernel<<<(G + 255) / 256, 256, 0, stream>>>(c, k, kp, out);
}